// MSLCrossEntropyLoss_3_57037165690976
// MI455X (gfx1250) — compile-verified
//
#include <hip/hip_runtime.h>

#define Cc 1000
#define Bn 65536
#define DECAYF 0.99f
#define ONE_M_DECAY 0.01f
#define ALPHAF 1.0f
#define LR_LF 0.05f
#define LR_MUF 0.02f
#define EPSF 1e-12f

typedef __attribute__((ext_vector_type(2))) float v2f;
typedef __attribute__((ext_vector_type(4))) float v4f;
typedef __attribute__((ext_vector_type(8))) float v8f;

// ---------------------------------------------------------------- zero scratch
__global__ void zero_ws(float* __restrict__ p, long n) {
    long i = (long)blockIdx.x * blockDim.x + threadIdx.x;
    long stride = (long)gridDim.x * blockDim.x;
    for (; i < n; i += stride) p[i] = 0.0f;
}

// ---------------------------------------------------------------- main pass
// One wave (32 lanes) per batch row; single streaming read of the 262MB logits
// via b128 loads. z' staged in LDS (8 rows x 1000 f32 = 32KB / 256-thr block),
// re-read twice from LDS with b128, never re-read from HBM.
__global__ __launch_bounds__(256) void main_pass(
    const float* __restrict__ logits, const int* __restrict__ targets,
    const float* __restrict__ Lam,
    float* __restrict__ seg, float* __restrict__ confc,
    float* __restrict__ priorc, float* __restrict__ cep)
{
    __shared__ float zbuf[8][Cc];
    const int wave = threadIdx.x >> 5;
    const int lane = threadIdx.x & 31;
    const int row  = blockIdx.x * 8 + wave;
    const int t    = targets[row];
    const float* __restrict__ lrow   = logits + (long)row * Cc;
    const float* __restrict__ lamrow = Lam + (long)t * Cc;
    const int NV = Cc / 4;                      // 250 float4 per row

    // pass 1: b128 loads, z' = z - Lam[t], argmax(z), max(z')
    float maxv = -3.4e38f; int maxi = 0x7fffffff;
    float maxzp = -3.4e38f;
    for (int j = lane; j < NV; j += 32) {
        const int col = j * 4;
        __builtin_prefetch(lrow + col + 512, 0, 3);     // global_prefetch_b8
        v4f z  = *(const v4f*)(lrow + col);
        v4f l  = *(const v4f*)(lamrow + col);
        v4f zp = z - l;
        *(v4f*)(&zbuf[wave][col]) = zp;                 // ds_store_b128
#pragma unroll
        for (int k = 0; k < 4; ++k) {
            if (z[k] > maxv) { maxv = z[k]; maxi = col + k; }  // first-index tie
            maxzp = fmaxf(maxzp, zp[k]);
        }
    }
    for (int off = 16; off; off >>= 1) {
        float ov = __shfl_xor(maxv, off, 32);
        int   oi = __shfl_xor(maxi, off, 32);
        if (ov > maxv || (ov == maxv && oi < maxi)) { maxv = ov; maxi = oi; }
        maxzp = fmaxf(maxzp, __shfl_xor(maxzp, off, 32));
    }
    __syncthreads();   // make LDS z' visible (waits dscnt)

    // pass 2 (LDS only, b128): sum exp
    float sum = 0.0f;
    for (int j = lane; j < NV; j += 32) {
        v4f zp = *(const v4f*)(&zbuf[wave][j * 4]);
        sum += __expf(zp[0] - maxzp) + __expf(zp[1] - maxzp)
             + __expf(zp[2] - maxzp) + __expf(zp[3] - maxzp);
    }
    for (int off = 16; off; off >>= 1) sum += __shfl_xor(sum, off, 32);
    const float lse = __logf(sum);
    const float inv = 1.0f / sum;

    if (lane == 0) {
        atomicAdd(&confc[(long)t * Cc + maxi], 1.0f);   // batch confusion counts
        atomicAdd(&priorc[t], 1.0f);                    // prior counts == cnt
        float logq_y = zbuf[wave][t] - maxzp - lse;     // CE contribution
        atomicAdd(&cep[row & 255], logq_y);             // spread partials
    }

    // pass 3 (LDS only, b128): q and scatter into segment sum
    float* __restrict__ segrow = seg + (long)t * Cc;
    for (int j = lane; j < NV; j += 32) {
        const int col = j * 4;
        v4f zp = *(const v4f*)(&zbuf[wave][col]);
        atomicAdd(&segrow[col + 0], __expf(zp[0] - maxzp) * inv);
        atomicAdd(&segrow[col + 1], __expf(zp[1] - maxzp) * inv);
        atomicAdd(&segrow[col + 2], __expf(zp[2] - maxzp) * inv);
        atomicAdd(&segrow[col + 3], __expf(zp[3] - maxzp) * inv);
    }
}

// ---------------------------------------------------------------- post rows
// One wave per 16 class-rows. Row sums via v_wmma_f32_16x16x4_f32 with an
// all-ones B matrix (B layout-invariant for ones): D[m][*] = rowsum(A[m][:]).
// A 16x4 f32 layout: lanes 0-15 hold K=0,1; lanes 16-31 hold K=2,3 (same M).
__global__ __launch_bounds__(32) void post_rows(
    const float* __restrict__ conf_ema, const float* __restrict__ Lam,
    const float* __restrict__ seg, const float* __restrict__ confc,
    const float* __restrict__ priorc,
    float* __restrict__ o_conf, float* __restrict__ o_lam)
{
    __shared__ float sCnt[16], sCountsSum[16], sEmpSum[16], sL1Sum[16];
    const int lane = threadIdx.x;
    const int m    = lane & 15;
    const int row  = blockIdx.x * 16 + m;
    const int k0   = (lane >> 4) << 1;       // 0 (lanes 0-15) or 2 (lanes 16-31)
    const bool rok = (row < Cc);
    const long base = (long)row * Cc;

    if (lane < 16) sCnt[m] = rok ? priorc[row] : 0.0f;

    const v2f bones = {1.0f, 1.0f};
    const v8f vzero = {0.f,0.f,0.f,0.f,0.f,0.f,0.f,0.f};

    // Pass A: conf_new = decay*ema + (1-decay)*counts; rowsum of (conf_new+1)*offdiag
    v8f acc = vzero;
    for (int c = 0; c < Cc; c += 4) {
        float a0 = 0.f, a1 = 0.f;
        if (rok) {
            int c0 = c + k0, c1 = c0 + 1;
            float cn0 = DECAYF * conf_ema[base + c0] + ONE_M_DECAY * confc[base + c0];
            float cn1 = DECAYF * conf_ema[base + c1] + ONE_M_DECAY * confc[base + c1];
            o_conf[base + c0] = cn0;
            o_conf[base + c1] = cn1;
            a0 = (c0 == row) ? 0.f : (cn0 + ALPHAF);
            a1 = (c1 == row) ? 0.f : (cn1 + ALPHAF);
        }
        v2f a = {a0, a1};
        acc = __builtin_amdgcn_wmma_f32_16x16x4_f32(false, a, false, bones,
                                                    (short)0, acc, false, false);
    }
    if (lane == 0)  for (int i = 0; i < 8; ++i) sCountsSum[i]     = acc[i];
    if (lane == 16) for (int i = 0; i < 8; ++i) sCountsSum[8 + i] = acc[i];
    __syncthreads();

    const float cnt  = sCnt[m];
    const float invc = 1.0f / fmaxf(cnt, 1.0f);

    // Pass B: rowsum of emp_raw = seg/max(cnt,1), diag zeroed
    acc = vzero;
    for (int c = 0; c < Cc; c += 4) {
        float a0 = 0.f, a1 = 0.f;
        if (rok) {
            int c0 = c + k0, c1 = c0 + 1;
            a0 = (c0 == row) ? 0.f : seg[base + c0] * invc;
            a1 = (c1 == row) ? 0.f : seg[base + c1] * invc;
        }
        v2f a = {a0, a1};
        acc = __builtin_amdgcn_wmma_f32_16x16x4_f32(false, a, false, bones,
                                                    (short)0, acc, false, false);
    }
    if (lane == 0)  for (int i = 0; i < 8; ++i) sEmpSum[i]     = acc[i];
    if (lane == 16) for (int i = 0; i < 8; ++i) sEmpSum[8 + i] = acc[i];
    __syncthreads();

    const float phiDen = 1.0f / (sCountsSum[m] + EPSF);       // phi: sum + eps
    const float empDen = 1.0f / fmaxf(sEmpSum[m], EPSF);      // emp: max(sum, eps)

    // Pass C: rowsum of L1 = (Lam + lr*(emp_n - phi)) * offdiag
    acc = vzero;
    for (int c = 0; c < Cc; c += 4) {
        float a0 = 0.f, a1 = 0.f;
        if (rok) {
            int c0 = c + k0, c1 = c0 + 1;
            if (c0 != row) {
                float cn  = DECAYF * conf_ema[base + c0] + ONE_M_DECAY * confc[base + c0];
                float phi = (cn + ALPHAF) * phiDen;
                float emp = seg[base + c0] * invc * empDen;
                a0 = Lam[base + c0] + LR_LF * (emp - phi);
            }
            if (c1 != row) {
                float cn  = DECAYF * conf_ema[base + c1] + ONE_M_DECAY * confc[base + c1];
                float phi = (cn + ALPHAF) * phiDen;
                float emp = seg[base + c1] * invc * empDen;
                a1 = Lam[base + c1] + LR_LF * (emp - phi);
            }
        }
        v2f a = {a0, a1};
        acc = __builtin_amdgcn_wmma_f32_16x16x4_f32(false, a, false, bones,
                                                    (short)0, acc, false, false);
    }
    if (lane == 0)  for (int i = 0; i < 8; ++i) sL1Sum[i]     = acc[i];
    if (lane == 16) for (int i = 0; i < 8; ++i) sL1Sum[8 + i] = acc[i];
    __syncthreads();

    const float rowMean = sL1Sum[m] * (1.0f / (float)Cc);
    const bool  present = (cnt > 0.0f);

    // Pass D: write Lam_new (row-centered L1 for present classes, else old Lam)
    for (int c = 0; c < Cc; c += 4) {
        if (!rok) continue;
        int c0 = c + k0, c1 = c0 + 1;
        float l0 = 0.f, l1 = 0.f;
        if (c0 != row) {
            float cn  = DECAYF * conf_ema[base + c0] + ONE_M_DECAY * confc[base + c0];
            float phi = (cn + ALPHAF) * phiDen;
            float emp = seg[base + c0] * invc * empDen;
            l0 = Lam[base + c0] + LR_LF * (emp - phi);
        }
        if (c1 != row) {
            float cn  = DECAYF * conf_ema[base + c1] + ONE_M_DECAY * confc[base + c1];
            float phi = (cn + ALPHAF) * phiDen;
            float emp = seg[base + c1] * invc * empDen;
            l1 = Lam[base + c1] + LR_LF * (emp - phi);
        }
        o_lam[base + c0] = present ? (l0 - rowMean) : Lam[base + c0];
        o_lam[base + c1] = present ? (l1 - rowMean) : Lam[base + c1];
    }
}

// ---------------------------------------------------------------- finalize
// Single block: prior_new + pi, emp_marg via column-sums of seg, mu update, CE.
__global__ __launch_bounds__(1024) void finalize(
    const float* __restrict__ prior_ema, const float* __restrict__ mu,
    const float* __restrict__ seg, const float* __restrict__ priorc,
    const float* __restrict__ cep,
    float* __restrict__ o_ce, float* __restrict__ o_mu, float* __restrict__ o_prior)
{
    __shared__ float red[1024];
    const int j = threadIdx.x;

    float pn = 0.0f;
    if (j < Cc) {
        pn = DECAYF * prior_ema[j] + ONE_M_DECAY * priorc[j];
        o_prior[j] = pn;
    }
    red[j] = pn; __syncthreads();
    for (int s = 512; s; s >>= 1) { if (j < s) red[j] += red[j + s]; __syncthreads(); }
    const float priorSum = red[0];
    __syncthreads();

    float mu1 = 0.0f;
    if (j < Cc) {
        float colsum = 0.0f;
        for (int r = 0; r < Cc; ++r) colsum += seg[(long)r * Cc + j];  // coalesced
        float emp_marg = colsum * (1.0f / (float)Bn);
        float pi = pn / (priorSum + EPSF);
        mu1 = mu[j] + LR_MUF * (emp_marg - pi);
    }
    red[j] = mu1; __syncthreads();
    for (int s = 512; s; s >>= 1) { if (j < s) red[j] += red[j + s]; __syncthreads(); }
    const float muMean = red[0] * (1.0f / (float)Cc);
    if (j < Cc) o_mu[j] = mu1 - muMean;

    if (j == 0) {
        float s = 0.0f;
        for (int i = 0; i < 256; ++i) s += cep[i];
        o_ce[0] = -s * (1.0f / (float)Bn);
    }
}

// ---------------------------------------------------------------- launch
extern "C" void kernel_launch(void* const* d_in, const int* in_sizes, int n_in,
                              void* d_out, int out_size, void* d_ws, size_t ws_size,
                              hipStream_t stream)
{
    const float* logits    = (const float*)d_in[0];
    const int*   targets   = (const int*)  d_in[1];
    const float* conf_ema  = (const float*)d_in[2];
    const float* prior_ema = (const float*)d_in[3];
    const float* Lam       = (const float*)d_in[4];
    const float* mu        = (const float*)d_in[5];

    float* out     = (float*)d_out;             // [ce | Lam_new | mu_new | conf_new | prior_new]
    float* o_ce    = out;
    float* o_lam   = out + 1;
    float* o_mu    = out + 1 + (long)Cc * Cc;
    float* o_conf  = out + 1 + (long)Cc * Cc + Cc;
    float* o_prior = out + 1 + 2L * Cc * Cc + Cc;

    float* ws     = (float*)d_ws;
    float* seg    = ws;                         // C*C segment sums of q
    float* confc  = ws + (long)Cc * Cc;         // C*C batch confusion counts
    float* priorc = ws + 2L * Cc * Cc;          // C target counts (== cnt)
    float* cep    = priorc + Cc;                // 256 CE partials

    const long nz = 2L * Cc * Cc + Cc + 256;
    zero_ws<<<1024, 256, 0, stream>>>(ws, nz);
    main_pass<<<Bn / 8, 256, 0, stream>>>(logits, targets, Lam, seg, confc, priorc, cep);
    post_rows<<<(Cc + 15) / 16, 32, 0, stream>>>(conf_ema, Lam, seg, confc, priorc,
                                                 o_conf, o_lam);
    finalize<<<1, 1024, 0, stream>>>(prior_ema, mu, seg, priorc, cep,
                                     o_ce, o_mu, o_prior);
}